// ResidualMLP_15814069584202
// MI455X (gfx1250) — compile-verified
//
#include <hip/hip_runtime.h>

// Problem constants (must match reference)
#define B_SZ    4096
#define MAXLEN  512
#define A_DIM   64
#define FEAT    256
#define XDIM    320   // FEAT + A_DIM
#define HID     512

typedef __attribute__((ext_vector_type(2))) float v2f;
typedef __attribute__((ext_vector_type(8))) float v8f;

// ---------------------------------------------------------------------------
// Kernel 1: action_base[b, :] = sum_{l < len[b]} padded[b, l, :] / max(len,1)
// One 256-thread block per batch row. float4-coalesced reads: full-speed HBM
// streaming (this step is the 512 MB / 23.3 TB/s ~ 22us roofline of the op).
// ---------------------------------------------------------------------------
__global__ __launch_bounds__(256) void masked_mean_kernel(
    const float* __restrict__ padded,
    const int*   __restrict__ lengths,
    float*       __restrict__ ab)        // (B_SZ, A_DIM)
{
    __shared__ float red[16][16][4];     // [row-phase][col4][component]
    const int b   = blockIdx.x;
    const int tid = threadIdx.x;
    const int c4  = tid & 15;            // float4 column group: cols 4*c4 .. 4*c4+3
    const int r   = tid >> 4;            // row phase 0..15
    const int len = lengths[b];

    const float4* p4 = (const float4*)(padded + (size_t)b * MAXLEN * A_DIM);
    float ax = 0.f, ay = 0.f, az = 0.f, aw = 0.f;
    for (int l = r; l < len; l += 16) {
        float4 v = p4[l * 16 + c4];
        ax += v.x; ay += v.y; az += v.z; aw += v.w;
    }
    red[r][c4][0] = ax; red[r][c4][1] = ay;
    red[r][c4][2] = az; red[r][c4][3] = aw;
    __syncthreads();

    for (int s = 8; s > 0; s >>= 1) {
        if (r < s) {
            red[r][c4][0] += red[r + s][c4][0];
            red[r][c4][1] += red[r + s][c4][1];
            red[r][c4][2] += red[r + s][c4][2];
            red[r][c4][3] += red[r + s][c4][3];
        }
        __syncthreads();
    }

    if (r == 0) {
        const float inv = 1.0f / (float)(len > 0 ? len : 1);
        float* dst = ab + (size_t)b * A_DIM + c4 * 4;
        dst[0] = red[0][c4][0] * inv;
        dst[1] = red[0][c4][1] * inv;
        dst[2] = red[0][c4][2] * inv;
        dst[3] = red[0][c4][3] * inv;
    }
}

// ---------------------------------------------------------------------------
// Kernel 2: fused MLP with fp32 WMMA (V_WMMA_F32_16X16X4_F32).
//   x = [concat_feat - cluster_center, action_base]   (16-row tile in LDS)
//   h = relu(x @ W1 + b1)                             (staged in LDS)
//   out = h @ W2 + b2 + action_base
// 256 threads = 8 wave32. fp32 WMMA keeps reference precision (W2 ~ 1e-4).
// Wave id is forced into an SGPR (readfirstlane) so all per-wave loops and
// guards are scalar branches: EXEC is structurally all-1s at every WMMA.
//
// Fragment layouts (ISA 7.12.2, wave32):
//   A 16x4 f32 : lane%16 = M ; lane<16 -> v={K0,K1}, lane>=16 -> v={K2,K3}
//   B  4x16 f32: lane%16 = N ; lane<16 -> v={K0,K1}, lane>=16 -> v={K2,K3}
//   C/D 16x16  : VGPR r, lane%16 = N, M = r + 8*(lane>=16)
// ---------------------------------------------------------------------------
__global__ __launch_bounds__(256) void fused_mlp_kernel(
    const float* __restrict__ cc,   // cluster_center (B, FEAT)
    const float* __restrict__ cf,   // concat_feat    (B, FEAT)
    const float* __restrict__ W1,   // (XDIM, HID)
    const float* __restrict__ b1,   // (HID)
    const float* __restrict__ W2,   // (HID, A_DIM)
    const float* __restrict__ b2,   // (A_DIM)
    const float* __restrict__ ab,   // action_base (B, A_DIM)
    float*       __restrict__ out)  // (B, A_DIM)
{
    constexpr int XS = XDIM + 1;    // 321: odd stride -> conflict-free col reads
    constexpr int HS = HID + 1;     // 513
    __shared__ float xs[16 * XS];   // 20544 B
    __shared__ float hs[16 * HS];   // 32832 B   (total 53376 B LDS)

    const int tid  = threadIdx.x;
    const int row0 = blockIdx.x * 16;

    // ---- Stage x tile: [delta | action_base] -------------------------------
    for (int idx = tid; idx < 16 * XDIM; idx += 256) {
        const int rr = idx / XDIM;
        const int c  = idx - rr * XDIM;
        const int gr = row0 + rr;
        float v;
        if (c < FEAT) v = cf[(size_t)gr * FEAT + c] - cc[(size_t)gr * FEAT + c];
        else          v = ab[(size_t)gr * A_DIM + (c - FEAT)];
        xs[rr * XS + c] = v;
    }
    __syncthreads();

    // Scalar wave id -> all per-wave control flow is SALU branches, EXEC
    // never masked around the WMMA regions.
    const int w    = __builtin_amdgcn_readfirstlane((int)(tid >> 5)); // 0..7
    const int lane = tid & 31;
    const int l15  = lane & 15;
    const int kh   = (lane >> 4) << 1;  // K sub-offset for this lane half: 0 or 2
    const int m8   = (lane >> 4) << 3;  // C/D row offset: 0 or 8

    // ---- GEMM1: h = relu(x @ W1 + b1), (16x320)*(320x512) ------------------
    for (int nt = w; nt < HID / 16; nt += 8) {
        v8f acc = {0.f, 0.f, 0.f, 0.f, 0.f, 0.f, 0.f, 0.f};
        const int n = nt * 16 + l15;
        #pragma unroll 8
        for (int k = 0; k < XDIM; k += 4) {
            v2f a, bm;
            a.x  = xs[l15 * XS + k + kh];
            a.y  = xs[l15 * XS + k + kh + 1];
            bm.x = W1[(size_t)(k + kh)     * HID + n];
            bm.y = W1[(size_t)(k + kh + 1) * HID + n];
            acc = __builtin_amdgcn_wmma_f32_16x16x4_f32(
                    false, a, false, bm, (short)0, acc, false, false);
        }
        const float bias = b1[n];
        #pragma unroll
        for (int r2 = 0; r2 < 8; ++r2) {
            const float h = acc[r2] + bias;
            hs[(m8 + r2) * HS + n] = h > 0.f ? h : 0.f;
        }
    }
    __syncthreads();

    // ---- GEMM2: out = h @ W2 + b2 + action_base, (16x512)*(512x64) ---------
    if (w < A_DIM / 16) {               // scalar guard: waves 0..3
        v8f acc = {0.f, 0.f, 0.f, 0.f, 0.f, 0.f, 0.f, 0.f};
        const int n = w * 16 + l15;
        #pragma unroll 8
        for (int k = 0; k < HID; k += 4) {
            v2f a, bm;
            a.x  = hs[l15 * HS + k + kh];
            a.y  = hs[l15 * HS + k + kh + 1];
            bm.x = W2[(size_t)(k + kh)     * A_DIM + n];
            bm.y = W2[(size_t)(k + kh + 1) * A_DIM + n];
            acc = __builtin_amdgcn_wmma_f32_16x16x4_f32(
                    false, a, false, bm, (short)0, acc, false, false);
        }
        const float bias = b2[n];
        #pragma unroll
        for (int r2 = 0; r2 < 8; ++r2) {
            const int m = m8 + r2;
            // residual action_base lives in the staged x tile at column FEAT+n
            out[(size_t)(row0 + m) * A_DIM + n] =
                acc[r2] + bias + xs[m * XS + FEAT + n];
        }
    }
}

// ---------------------------------------------------------------------------
extern "C" void kernel_launch(void* const* d_in, const int* in_sizes, int n_in,
                              void* d_out, int out_size, void* d_ws, size_t ws_size,
                              hipStream_t stream)
{
    (void)in_sizes; (void)n_in; (void)out_size; (void)ws_size;
    const float* padded  = (const float*)d_in[0];
    const int*   lengths = (const int*)  d_in[1];
    const float* cc      = (const float*)d_in[2];
    const float* cf      = (const float*)d_in[3];
    const float* W1      = (const float*)d_in[4];
    const float* b1      = (const float*)d_in[5];
    const float* W2      = (const float*)d_in[6];
    const float* b2      = (const float*)d_in[7];
    float* out = (float*)d_out;
    float* ab  = (float*)d_ws;           // needs B_SZ*A_DIM*4 = 1 MiB scratch

    masked_mean_kernel<<<B_SZ, 256, 0, stream>>>(padded, lengths, ab);
    fused_mlp_kernel<<<B_SZ / 16, 256, 0, stream>>>(cc, cf, W1, b1, W2, b2, ab, out);
}